// CornerPooling_89593017794766
// MI455X (gfx1250) — compile-verified
//
#include <hip/hip_runtime.h>

// CornerPooling on MI455X (gfx1250): memory-bound fp32 stencil/scan chain.
// 3x3 convs computed EXACTLY via V_WMMA_F32_16X16X4_F32 (banded vertical-tap
// matrix x image tile, summed over the 3 kernel columns). Tiles are staged
// into LDS with gfx1250 async-to-LDS loads (ASYNCcnt) — 5x b128 per lane —
// then consumed branch-free from LDS with cndmask zero-padding.
// Workspace layout (floats): [0..63] const block, then two full images.

typedef __attribute__((ext_vector_type(2))) float v2f;
typedef __attribute__((ext_vector_type(8))) float v8f;

#define IMG_H 1024
#define IMG_W 1024
#define IMG_B 32

#define LDS_ROWS 20     // rows h0-1 .. h0+18 (chunk c=4,k=3 -> r=19)
#define LDS_PITCH 40    // floats; half-waves land on disjoint bank ranges

// ---------------------------------------------------------------------------
// Fold BN params into effective conv weights/biases (device-resident scalars).
// Const block layout:
//  [0..8]  wa*sa, [9]  (ba-ma)*sa+bta          (conv3x3_bn_r, used twice)
//  [10..18] wb*sb, [19] (bb-mb)*sb+btb          (conv3x3_bn)
//  [20] wc*sc, [21] (bc-mc)*sc+btc              (conv1x1_bn branch)
//  [22..30] wd, [31] bd                         (conv3x3_r)
//  [32] we, [33] be                             (final conv1x1)
// ---------------------------------------------------------------------------
__global__ void cp_prep_consts(
    const float* wa, const float* ba, const float* ga, const float* bta,
    const float* ma, const float* va,
    const float* wb, const float* bb, const float* gb, const float* btb,
    const float* mb, const float* vb,
    const float* wc, const float* bc, const float* gc, const float* btc,
    const float* mc, const float* vc,
    const float* wd, const float* bd, const float* we, const float* be,
    float* consts)
{
    if (threadIdx.x == 0 && blockIdx.x == 0) {
        const float EPS = 1e-5f;
        float sa = ga[0] * rsqrtf(va[0] + EPS);
        for (int j = 0; j < 9; ++j) consts[j] = wa[j] * sa;
        consts[9] = (ba[0] - ma[0]) * sa + bta[0];
        float sb = gb[0] * rsqrtf(vb[0] + EPS);
        for (int j = 0; j < 9; ++j) consts[10 + j] = wb[j] * sb;
        consts[19] = (bb[0] - mb[0]) * sb + btb[0];
        float sc = gc[0] * rsqrtf(vc[0] + EPS);
        consts[20] = wc[0] * sc;
        consts[21] = (bc[0] - mc[0]) * sc + btc[0];
        for (int j = 0; j < 9; ++j) consts[22 + j] = wd[j];
        consts[31] = bd[0];
        consts[32] = we[0];
        consts[33] = be[0];
    }
}

// ---------------------------------------------------------------------------
// Generic fused 3x3 conv via WMMA. One wave = one 16x16 output tile.
//   acc(16x16) = sum_{dx=-1..1} sum_{c=0..4} A(dx,c) x B(dx,c)   (K=4 chunks)
// A[m][k] = kv_dx[4c+k-m] (banded, from the 3 vertical taps of column dx),
// B[k][n] = in[h0-1+4c+k][w0+n+dx] (zero outside image = SAME padding).
// Lane->K mapping k = j + 2*(lane>=16) identical for A and B (permutation-safe).
// Input tile staged via async-to-LDS b128 loads; consumed branch-free.
// All masking is done with flat selects (v_cndmask) — no EXEC manipulation,
// keeping EXEC all-ones as WMMA requires and avoiding branch/SALU overhead.
// ---------------------------------------------------------------------------
__global__ __launch_bounds__(256) void cp_conv3x3_wmma(
    const float* __restrict__ in, const float* __restrict__ aux,
    float* __restrict__ out, const float* __restrict__ consts,
    int wo, int auxo, int posto, int do_relu)
{
    __shared__ float smem[8 * LDS_ROWS * LDS_PITCH];   // 25.6 KB / block

    const int lane = threadIdx.x & 31;
    const int wib  = threadIdx.x >> 5;                 // wave in block
    const int wid  = blockIdx.x * (blockDim.x >> 5) + wib;

    const int tilesPerImg = (IMG_H / 16) * (IMG_W / 16);   // 4096
    const int b  = wid >> 12;
    const int r0 = wid & (tilesPerImg - 1);
    const int h0 = (r0 >> 6) << 4;
    const int w0 = (r0 & 63) << 4;

    const float* img = in + (size_t)b * ((size_t)IMG_H * IMG_W);
    float* swave = smem + wib * (LDS_ROWS * LDS_PITCH);

    // ---- Stage 20x32 halo window into LDS with async b128 loads -----------
    // Window columns [wstart, wstart+31]; wstart 16B-aligned, covers w0-1..w0+16.
    const int wstart = min(max(w0 - 4, 0), IMG_W - 32);
    {
        const int rsub = lane >> 3;     // 0..3 : row within group of 4
        const int cq   = lane & 7;      // 0..7 : 4-float column quad
        const unsigned ldsBase = (unsigned)(size_t)swave;
#pragma unroll
        for (int it = 0; it < 5; ++it) {
            const int r    = it * 4 + rsub;                          // 0..19
            const int grow = min(max(h0 - 1 + r, 0), IMG_H - 1);     // clamp
            const float* gp = img + (size_t)grow * IMG_W + (wstart + cq * 4);
            const unsigned la = ldsBase + (unsigned)((r * LDS_PITCH + cq * 4) * 4);
            asm volatile("global_load_async_to_lds_b128 %0, %1, off"
                         :: "v"(la), "v"((unsigned long long)(size_t)gp)
                         : "memory");
        }
        asm volatile("s_wait_asynccnt 0" ::: "memory");
        // wave-private region: no cross-wave barrier needed
    }

    // ---- Effective weights / bias -----------------------------------------
    float wk[9];
#pragma unroll
    for (int j = 0; j < 9; ++j) wk[j] = consts[wo + j];
    const float bias = consts[wo + 9];

    const int n    = lane & 15;   // B/D column within tile
    const int half = lane >> 4;   // selects K pair and D row half
    const int m_a  = lane & 15;   // A row

    v8f acc;
#pragma unroll
    for (int v = 0; v < 8; ++v) acc[v] = bias;

#pragma unroll
    for (int dx = -1; dx <= 1; ++dx) {
        const int  cw   = w0 + n + dx;
        const bool wok  = ((unsigned)cw < (unsigned)IMG_W);
        const int  cl   = min(max(cw - wstart, 0), 31);   // safe LDS column
        const float kv0 = wk[0 * 3 + (dx + 1)];
        const float kv1 = wk[1 * 3 + (dx + 1)];
        const float kv2 = wk[2 * 3 + (dx + 1)];
#pragma unroll
        for (int c = 0; c < 5; ++c) {
            v2f Bv, Av;
#pragma unroll
            for (int j = 0; j < 2; ++j) {
                const int k = j + 2 * half;
                const int r = 4 * c + k;              // LDS row 0..19
                const int i = h0 - 1 + r;             // global row
                const float raw = swave[r * LDS_PITCH + cl];
                const bool ok = wok & ((unsigned)i < (unsigned)IMG_H);
                Bv[j] = ok ? raw : 0.0f;              // v_cndmask zero pad
                // A operand: banded vertical-tap matrix, flat select chain
                const int d = r - m_a;
                float aval = (d == 0) ? kv0 : 0.0f;   // v_cndmask
                aval       = (d == 1) ? kv1 : aval;   // v_cndmask
                aval       = (d == 2) ? kv2 : aval;   // v_cndmask
                Av[j] = aval;
            }
            acc = __builtin_amdgcn_wmma_f32_16x16x4_f32(
                /*neg_a=*/false, Av, /*neg_b=*/false, Bv,
                /*c_mod=*/(short)0, acc, /*reuse_a=*/false, /*reuse_b=*/false);
        }
    }

    // ---- Epilogue ---------------------------------------------------------
    float c1 = 0.0f, c0 = 0.0f;
    if (auxo >= 0) { c1 = consts[auxo]; c0 = consts[auxo + 1]; }
    float pm = 1.0f, pb = 0.0f;
    if (posto >= 0) { pm = consts[posto]; pb = consts[posto + 1]; }

    float* outp = out + (size_t)b * ((size_t)IMG_H * IMG_W);
    const float* auxp =
        (auxo >= 0) ? (aux + (size_t)b * ((size_t)IMG_H * IMG_W)) : nullptr;

#pragma unroll
    for (int v = 0; v < 8; ++v) {
        const int row = h0 + v + 8 * half;    // D layout: VGPR v -> M = v+8*half
        const int col = w0 + n;               // lane&15 -> N
        const size_t idx = (size_t)row * IMG_W + col;
        float val = acc[v];
        if (auxo >= 0) val += auxp[idx] * c1 + c0;
        if (do_relu)   val = fmaxf(val, 0.0f);
        val = pm * val + pb;
        outp[idx] = val;
    }
}

// ---------------------------------------------------------------------------
// Reverse (right->left) inclusive cummax along W. One 256-thread block per row,
// 4 elements per thread; LDS Hillis-Steele suffix-max across threads.
// ---------------------------------------------------------------------------
__global__ __launch_bounds__(256) void cp_row_revcummax(
    const float* __restrict__ in, float* __restrict__ out)
{
    __shared__ float lds[256];
    const int t = threadIdx.x;
    const size_t row = blockIdx.x;              // b*H + h
    const float* p = in + row * IMG_W + (size_t)t * 4;

    float x0 = p[0], x1 = p[1], x2 = p[2], x3 = p[3];
    float s3 = x3;
    float s2 = fmaxf(x2, s3);
    float s1 = fmaxf(x1, s2);
    float s0 = fmaxf(x0, s1);

    lds[t] = s0;
    __syncthreads();
    float m = s0;
#pragma unroll
    for (int off = 1; off < 256; off <<= 1) {
        float v = (t + off < 256) ? lds[t + off] : -__builtin_inff();
        __syncthreads();
        m = fmaxf(m, v);
        lds[t] = m;
        __syncthreads();
    }
    const float suf = (t < 255) ? lds[t + 1] : -__builtin_inff();

    float* q = out + row * IMG_W + (size_t)t * 4;
    q[0] = fmaxf(s0, suf);
    q[1] = fmaxf(s1, suf);
    q[2] = fmaxf(s2, suf);
    q[3] = fmaxf(s3, suf);
}

// ---------------------------------------------------------------------------
// Reverse (bottom->top) cummax along H, fused with i12 = i1 + i2.
// One thread per (b, w) column; loads coalesced across lanes. In-place on i1.
// ---------------------------------------------------------------------------
__global__ __launch_bounds__(256) void cp_col_revcummax_add(
    const float* __restrict__ x1buf, float* __restrict__ i1buf)
{
    const int g = blockIdx.x * blockDim.x + threadIdx.x;   // 0 .. B*W-1
    const int b = g >> 10;
    const int w = g & (IMG_W - 1);
    const float* xp = x1buf + (size_t)b * ((size_t)IMG_H * IMG_W) + w;
    float*       ip = i1buf + (size_t)b * ((size_t)IMG_H * IMG_W) + w;
    float run = -__builtin_inff();
#pragma unroll 4
    for (int h = IMG_H - 1; h >= 0; --h) {
        const size_t off = (size_t)h * IMG_W;
        run = fmaxf(run, xp[off]);
        ip[off] = ip[off] + run;
    }
}

// ---------------------------------------------------------------------------
extern "C" void kernel_launch(void* const* d_in, const int* in_sizes, int n_in,
                              void* d_out, int out_size, void* d_ws, size_t ws_size,
                              hipStream_t stream)
{
    const float* x   = (const float*)d_in[0];
    const float* wa  = (const float*)d_in[1];
    const float* ba  = (const float*)d_in[2];
    const float* ga  = (const float*)d_in[3];
    const float* bta = (const float*)d_in[4];
    const float* ma  = (const float*)d_in[5];
    const float* va  = (const float*)d_in[6];
    const float* wb  = (const float*)d_in[7];
    const float* bb  = (const float*)d_in[8];
    const float* gb  = (const float*)d_in[9];
    const float* btb = (const float*)d_in[10];
    const float* mb  = (const float*)d_in[11];
    const float* vb  = (const float*)d_in[12];
    const float* wc  = (const float*)d_in[13];
    const float* bc  = (const float*)d_in[14];
    const float* gc  = (const float*)d_in[15];
    const float* btc = (const float*)d_in[16];
    const float* mc  = (const float*)d_in[17];
    const float* vc  = (const float*)d_in[18];
    const float* wd  = (const float*)d_in[19];
    const float* bd  = (const float*)d_in[20];
    const float* we  = (const float*)d_in[21];
    const float* be  = (const float*)d_in[22];

    const size_t N = (size_t)IMG_B * IMG_H * IMG_W;   // 33,554,432 elements
    float* consts = (float*)d_ws;                     // 64 floats
    float* buf0   = consts + 64;                      // image-sized scratch A
    float* buf1   = buf0 + N;                         // image-sized scratch B
    float* outp   = (float*)d_out;

    cp_prep_consts<<<1, 1, 0, stream>>>(wa, ba, ga, bta, ma, va,
                                        wb, bb, gb, btb, mb, vb,
                                        wc, bc, gc, btc, mc, vc,
                                        wd, bd, we, be, consts);

    const int wavesTotal  = IMG_B * (IMG_H / 16) * (IMG_W / 16); // 131072 tiles
    const int convBlocks  = wavesTotal / 8;                      // 8 waves/block

    // x1 = relu(bn(conv3(x, wa)))            -> buf0
    cp_conv3x3_wmma<<<convBlocks, 256, 0, stream>>>(
        x, nullptr, buf0, consts, /*wo=*/0, /*auxo=*/-1, /*posto=*/-1, /*relu=*/1);

    // i1 = revcummax_W(x1)                   -> buf1
    cp_row_revcummax<<<IMG_B * IMG_H, 256, 0, stream>>>(buf0, buf1);

    // i12 = i1 + revcummax_H(x1)             -> buf1 (in place)
    cp_col_revcummax_add<<<(IMG_B * IMG_W) / 256, 256, 0, stream>>>(buf0, buf1);

    // s = relu(bn(conv3(i12, wb)) + bn(conv1x1(x, wc)))   -> buf0
    cp_conv3x3_wmma<<<convBlocks, 256, 0, stream>>>(
        buf1, x, buf0, consts, /*wo=*/10, /*auxo=*/20, /*posto=*/-1, /*relu=*/1);

    // t = relu(bn(conv3(s, wa)))             -> buf1   (shared module)
    cp_conv3x3_wmma<<<convBlocks, 256, 0, stream>>>(
        buf0, nullptr, buf1, consts, /*wo=*/0, /*auxo=*/-1, /*posto=*/-1, /*relu=*/1);

    // out = we * relu(conv3(t, wd) + bd) + be -> d_out
    cp_conv3x3_wmma<<<convBlocks, 256, 0, stream>>>(
        buf1, nullptr, outp, consts, /*wo=*/22, /*auxo=*/-1, /*posto=*/32, /*relu=*/1);

    (void)in_sizes; (void)n_in; (void)out_size; (void)ws_size;
}